// Detection_19868518711724
// MI455X (gfx1250) — compile-verified
//
#include <hip/hip_runtime.h>
#include <math.h>

#define BSZ 8
#define PP  8192
#define GG  16
#define DD  512

typedef float v2f __attribute__((ext_vector_type(2)));
typedef float v8f __attribute__((ext_vector_type(8)));

#define WSTRIDE 516   // LDS stride (floats) for weight columns: conflict-free b64 reads
#define CSTRIDE 17    // LDS stride for C-tile dump
#define WAVES   8     // waves per block (256 threads)

__device__ __forceinline__ float sl1(float d) {
    float a = fabsf(d);
    return (a < 1.0f) ? 0.5f * d * d : a - 0.5f;
}

__global__ __launch_bounds__(256) void det_main(
    const float* __restrict__ v,
    const float* __restrict__ gt,
    const float* __restrict__ rois,
    const float* __restrict__ labels,
    const float* __restrict__ pre_score,
    const float* __restrict__ cls_w,
    const float* __restrict__ cls_b,
    const float* __restrict__ reg_w,
    const float* __restrict__ reg_b,
    float* __restrict__ ws)
{
    __shared__ float lds_w[16 * WSTRIDE];            // B matrix columns (N=0..15), cols 8..15 zero
    __shared__ float lds_c[WAVES * 16 * CSTRIDE];    // per-wave C tile dump

    const int tid = threadIdx.x;

    // zero whole weight region (cols 8..15 must be 0)
    for (int i = tid; i < 16 * WSTRIDE; i += 256) lds_w[i] = 0.0f;
    __syncthreads();
    // fill cols 0..3 = cls_w rows, cols 4..7 = reg_w rows (both stored k-contiguous)
    for (int i = tid; i < 8 * DD; i += 256) {
        int n = i >> 9;          // / 512
        int k = i & (DD - 1);
        float val = (n < 4) ? cls_w[n * DD + k] : reg_w[(n - 4) * DD + k];
        lds_w[n * WSTRIDE + k] = val;
    }
    __syncthreads();

    const int wave = tid >> 5;
    const int lane = tid & 31;
    const int tile = blockIdx.x * WAVES + wave;      // 16-row tile index over BS*P rows
    const int mrow = lane & 15;
    const int koff = (lane >> 4) << 1;               // lanes 0-15 -> K 0,1 ; lanes 16-31 -> K 2,3

    const int row = tile * 16 + mrow;                // global row in v (= b*P + p)
    const float* aptr = v + (size_t)row * DD + koff;
    const float* bptr = lds_w + mrow * WSTRIDE + koff;

    v8f acc = {};
#pragma unroll 4
    for (int k = 0; k < DD; k += 4) {
        v2f a, bb;
        a.x  = aptr[k];
        a.y  = aptr[k + 1];
        bb.x = bptr[k];
        bb.y = bptr[k + 1];
        // D = A(16x4 f32) * B(4x16 f32) + C  -> v_wmma_f32_16x16x4_f32
        acc = __builtin_amdgcn_wmma_f32_16x16x4_f32(
            /*neg_a=*/false, a, /*neg_b=*/false, bb,
            /*c_mod=*/(short)0, acc, /*reuse_a=*/false, /*reuse_b=*/false);
    }

    // bias for this output column (n = mrow): cols 0-3 cls_b, 4-7 reg_b, 8-15 unused
    float bias = 0.0f;
    if (mrow < 4)      bias = cls_b[mrow];
    else if (mrow < 8) bias = reg_b[mrow - 4];
#pragma unroll
    for (int r = 0; r < 8; ++r) acc[r] += bias;

    // dump C tile: element (m, n): m = r + (lane>=16 ? 8 : 0), n = mrow
    {
        float* cbase = lds_c + wave * 16 * CSTRIDE;
        const int mbase = (lane >> 4) << 3;
#pragma unroll
        for (int r = 0; r < 8; ++r)
            cbase[(mbase + r) * CSTRIDE + mrow] = acc[r];
    }
    __syncthreads();

    // ---- per-roi loss: lanes 0..15 each handle one row of the tile ----
    const int b = tile / (PP / 16);
    float cnum = 0.0f, ccnt = 0.0f, l1c = 0.0f;
    if (lane < 16) {
        const float* crow = lds_c + wave * 16 * CSTRIDE + lane * CSTRIDE;
        float c0 = crow[0], c1 = crow[1], c2 = crow[2], c3 = crow[3];
        float rg0 = crow[4], rg1 = crow[5], rg2 = crow[6], rg3 = crow[7];

        // clip then log-softmax CE
        const float lo = 1e-7f, hi = 1.0f - 1e-7f;
        c0 = fminf(fmaxf(c0, lo), hi);
        c1 = fminf(fmaxf(c1, lo), hi);
        c2 = fminf(fmaxf(c2, lo), hi);
        c3 = fminf(fmaxf(c3, lo), hi);
        float mx  = fmaxf(fmaxf(c0, c1), fmaxf(c2, c3));
        float lse = mx + logf(expf(c0 - mx) + expf(c1 - mx) +
                              expf(c2 - mx) + expf(c3 - mx));

        // label = argmax(labels[b]) (first max)
        const float* lb = labels + b * 4;
        int lab = 0; float lm = lb[0];
#pragma unroll
        for (int j = 1; j < 4; ++j) { float t = lb[j]; if (t > lm) { lm = t; lab = j; } }
        float clab = (lab == 0) ? c0 : (lab == 1) ? c1 : (lab == 2) ? c2 : c3;
        float ce = lse - clab;

        const int prow = tile * 16 + lane;           // = b*P + p_local
        float ps = pre_score[(size_t)prow * 4 + lab];

        const float4 rr = *(const float4*)(rois + (size_t)prow * 4);
        float area_r = (rr.z - rr.x) * (rr.w - rr.y);

        float best = -1.0f;
        float4 gbest = make_float4(0.f, 0.f, 0.f, 0.f);
        float anym = 0.0f;
#pragma unroll
        for (int g = 0; g < GG; ++g) {
            float4 gb = *(const float4*)(gt + (size_t)(b * GG + g) * 4);
            float ltx = fmaxf(gb.x, rr.x), lty = fmaxf(gb.y, rr.y);
            float rbx = fminf(gb.z, rr.z), rby = fminf(gb.w, rr.w);
            float w = fmaxf(rbx - ltx, 0.0f), h = fmaxf(rby - lty, 0.0f);
            float inter = w * h;
            float ag = (gb.z - gb.x) * (gb.w - gb.y);
            float iou = inter / (ag + area_r - inter);
            if (iou > 0.5f) anym = 1.0f;
            if (iou > best) { best = iou; gbest = gb; }
        }

        // regression targets
        float gx = (gbest.z + gbest.x) * 0.5f, gy = (gbest.w + gbest.y) * 0.5f;
        float gw = (gbest.z - gbest.x) * 0.5f, gh = (gbest.w - gbest.y) * 0.5f;
        float rx = (rr.z + rr.x) * 0.5f, ry = (rr.w + rr.y) * 0.5f;
        float rw = (rr.z - rr.x) * 0.5f, rh = (rr.w - rr.y) * 0.5f;
        float tx = (gx - rx) / (rw + 1e-8f);
        float ty = (gy - ry) / (rh + 1e-8f);
        float tw = logf(gw / (rw + 1e-8f));
        float th = logf(gh / (rh + 1e-8f));
        float l1 = sl1(rg0 - tx) + sl1(rg1 - ty) + sl1(rg2 - tw) + sl1(rg3 - th);

        cnum = ce * ps * anym;
        ccnt = anym;
        l1c  = l1 * anym * ps;
    }

    // wave reduction (lanes >=16 contribute zeros)
#pragma unroll
    for (int off = 16; off; off >>= 1) {
        cnum += __shfl_xor(cnum, off, 32);
        ccnt += __shfl_xor(ccnt, off, 32);
        l1c  += __shfl_xor(l1c,  off, 32);
    }
    if (lane == 0) {
        atomicAdd(&ws[b * 3 + 0], cnum);
        atomicAdd(&ws[b * 3 + 1], ccnt);
        atomicAdd(&ws[b * 3 + 2], l1c);
    }
}

__global__ void det_final(const float* __restrict__ ws, float* __restrict__ out) {
    if (threadIdx.x == 0 && blockIdx.x == 0) {
        float accc = 0.0f, accl = 0.0f;
#pragma unroll
        for (int b = 0; b < BSZ; ++b) {
            accc += ws[b * 3 + 0] / (ws[b * 3 + 1] + 1e-7f);
            accl += ws[b * 3 + 2];
        }
        out[0] = accc / ((float)BSZ + 1e-7f) + accl / (float)(BSZ * PP);
    }
}

extern "C" void kernel_launch(void* const* d_in, const int* in_sizes, int n_in,
                              void* d_out, int out_size, void* d_ws, size_t ws_size,
                              hipStream_t stream) {
    (void)in_sizes; (void)n_in; (void)out_size; (void)ws_size;
    const float* v         = (const float*)d_in[0];
    const float* gt        = (const float*)d_in[1];
    const float* rois      = (const float*)d_in[2];
    const float* labels    = (const float*)d_in[3];
    const float* pre_score = (const float*)d_in[4];
    const float* cls_w     = (const float*)d_in[5];
    const float* cls_b     = (const float*)d_in[6];
    const float* reg_w     = (const float*)d_in[7];
    const float* reg_b     = (const float*)d_in[8];
    float* ws = (float*)d_ws;

    hipMemsetAsync(ws, 0, BSZ * 3 * sizeof(float), stream);

    const int tiles  = (BSZ * PP) / 16;   // 4096 16-row tiles
    const int blocks = tiles / WAVES;     // 512 blocks of 8 waves
    det_main<<<blocks, 256, 0, stream>>>(v, gt, rois, labels, pre_score,
                                         cls_w, cls_b, reg_w, reg_b, ws);
    det_final<<<1, 32, 0, stream>>>(ws, (float*)d_out);
}